// CombinedLoss_14216341749925
// MI455X (gfx1250) — compile-verified
//
#include <hip/hip_runtime.h>
#include <hip/hip_bf16.h>
#include <math.h>
#include <stdint.h>

#define WAVES_PER_BLOCK 8
#define NBLOCKS 512
#define TOTAL_WAVES (WAVES_PER_BLOCK * NBLOCKS)
#define NCLASS 1024
#define ROW_BYTES (NCLASS * 4)

// Issue one 4KB row (1024 fp32) as 8 async b128 loads into LDS.
// IOFFSET is added to BOTH the LDS destination and the global source address
// (CDNA5 ISA), so a single (lds, global) base pair per lane
// plus immediate offsets covers the whole row: lane*16 + k*512, k=0..7.
// TH_LOAD_NT: 256MB read-once stream must not thrash WGP$/L2.
__device__ __forceinline__ void issue_async_row(unsigned lds_base,
                                                const float* __restrict__ grow,
                                                int lane) {
  const char* g = (const char*)grow + lane * 16;
  unsigned l = lds_base + (unsigned)(lane * 16);
  asm volatile(
      "global_load_async_to_lds_b128 %0, %1, off th:TH_LOAD_NT\n\t"
      "global_load_async_to_lds_b128 %0, %1, off offset:512 th:TH_LOAD_NT\n\t"
      "global_load_async_to_lds_b128 %0, %1, off offset:1024 th:TH_LOAD_NT\n\t"
      "global_load_async_to_lds_b128 %0, %1, off offset:1536 th:TH_LOAD_NT\n\t"
      "global_load_async_to_lds_b128 %0, %1, off offset:2048 th:TH_LOAD_NT\n\t"
      "global_load_async_to_lds_b128 %0, %1, off offset:2560 th:TH_LOAD_NT\n\t"
      "global_load_async_to_lds_b128 %0, %1, off offset:3072 th:TH_LOAD_NT\n\t"
      "global_load_async_to_lds_b128 %0, %1, off offset:3584 th:TH_LOAD_NT"
      :: "v"(l), "v"(g) : "memory");
}

// Consume one row resident in LDS: argmax + logsumexp + per-row loss term.
__device__ __forceinline__ float process_row(const float* __restrict__ buf,
                                             int lane, int r,
                                             const int* __restrict__ truev,
                                             const float* __restrict__ centers,
                                             float invB) {
  const float4* b4 = (const float4*)buf;
  float4 q[8];
#pragma unroll
  for (int k = 0; k < 8; ++k) q[k] = b4[k * 32 + lane];  // ds_load_b128

  // In-lane max/argmax; strict '>' + increasing column order => first occurrence.
  float m = -INFINITY;
  int mi = 0;
#pragma unroll
  for (int k = 0; k < 8; ++k) {
    int base = 4 * (k * 32 + lane);
    if (q[k].x > m) { m = q[k].x; mi = base; }
    if (q[k].y > m) { m = q[k].y; mi = base + 1; }
    if (q[k].z > m) { m = q[k].z; mi = base + 2; }
    if (q[k].w > m) { m = q[k].w; mi = base + 3; }
  }
  // Wave32 reduction; tie -> smaller index (jnp.argmax semantics).
#pragma unroll
  for (int off = 16; off > 0; off >>= 1) {
    float om = __shfl_xor(m, off, 32);
    int omi = __shfl_xor(mi, off, 32);
    if (om > m || (om == m && omi < mi)) { m = om; mi = omi; }
  }

  float s = 0.f;
#pragma unroll
  for (int k = 0; k < 8; ++k) {
    s += __expf(q[k].x - m) + __expf(q[k].y - m) +
         __expf(q[k].z - m) + __expf(q[k].w - m);
  }
#pragma unroll
  for (int off = 16; off > 0; off >>= 1) s += __shfl_xor(s, off, 32);

  float loss = 0.f;
  if (lane == 0) {
    int t = truev[r];
    float pt = buf[t];                       // row still in LDS
    float nll = (m - pt) + __logf(s);        // max + log(sum exp) - pred[true]
    float ax = centers[2 * mi], ay = centers[2 * mi + 1];
    float tx = centers[2 * t], ty = centers[2 * t + 1];
    float dx = ax - tx, dy = ay - ty;
    float d2 = dx * dx + dy * dy;
    float d = (d2 > 0.f) ? sqrtf(d2) : 0.f;
    loss = 0.5f * d * (1.0f / 255.0f) + 0.5f * nll * invB;
  }
  return loss;
}

__global__ __launch_bounds__(256) void fused_loss_kernel(
    const float* __restrict__ pred, const int* __restrict__ truev,
    const float* __restrict__ centers, float* __restrict__ ws,
    int B, int rows_per_wave, float invB) {
  __shared__ __align__(16) float smem[WAVES_PER_BLOCK * 2 * NCLASS];  // 64 KB
  const int lane = threadIdx.x & 31;
  const int w = threadIdx.x >> 5;
  const int gw = blockIdx.x * WAVES_PER_BLOCK + w;
  const long r0 = (long)gw * rows_per_wave;

  const unsigned lds_base = (unsigned)(size_t)(&smem[(w * 2) * NCLASS]);

  float acc = 0.f;
  if (r0 < B) {
    issue_async_row(lds_base, pred + r0 * NCLASS, lane);
    for (int it = 0; it < rows_per_wave; ++it) {
      long r = r0 + it;
      if (r >= B) break;
      const int b = it & 1;
      const bool pref = (it + 1 < rows_per_wave) && (r + 1 < B);
      if (pref) {
        // Previous use of the target buffer must be fully read out of LDS
        // before the async engine overwrites it.
        asm volatile("s_wait_dscnt 0x0" ::: "memory");
        issue_async_row(lds_base + (unsigned)(((it + 1) & 1) * ROW_BYTES),
                        pred + (r + 1) * NCLASS, lane);
        // Asyncs retire in order: <=8 outstanding => buffer 'b' has landed.
        asm volatile("s_wait_asynccnt 0x8" ::: "memory");
      } else {
        asm volatile("s_wait_asynccnt 0x0" ::: "memory");
      }
      acc += process_row(&smem[(w * 2 + b) * NCLASS], lane, (int)r,
                         truev, centers, invB);
    }
  }
  if (lane == 0) ws[gw] = acc;  // always written -> no stale state across calls
}

__global__ __launch_bounds__(256) void reduce_partials(
    const float* __restrict__ ws, float* __restrict__ out, int n) {
  __shared__ float red[256];
  float s = 0.f;
  for (int i = threadIdx.x; i < n; i += 256) s += ws[i];
  red[threadIdx.x] = s;
  __syncthreads();
#pragma unroll
  for (int st = 128; st > 0; st >>= 1) {
    if ((int)threadIdx.x < st) red[threadIdx.x] += red[threadIdx.x + st];
    __syncthreads();
  }
  if (threadIdx.x == 0) out[0] = red[0];  // deterministic fixed-order tree
}

extern "C" void kernel_launch(void* const* d_in, const int* in_sizes, int n_in,
                              void* d_out, int out_size, void* d_ws, size_t ws_size,
                              hipStream_t stream) {
  const float* pred = (const float*)d_in[0];
  const int* truev = (const int*)d_in[1];     // jnp int64 -> int32 per harness
  const float* centers = (const float*)d_in[2];
  float* out = (float*)d_out;
  float* ws = (float*)d_ws;                    // TOTAL_WAVES floats = 16 KB

  const int B = in_sizes[1];
  const int rows_per_wave = (B + TOTAL_WAVES - 1) / TOTAL_WAVES;

  fused_loss_kernel<<<NBLOCKS, 256, 0, stream>>>(pred, truev, centers, ws,
                                                 B, rows_per_wave, 1.0f / (float)B);
  reduce_partials<<<1, 256, 0, stream>>>(ws, out, TOTAL_WAVES);
}